// OptimizedEMDHybridInSARModel_85779086835977
// MI455X (gfx1250) — compile-verified
//
#include <hip/hip_runtime.h>
#include <math.h>

// Problem constants from the reference
#define NN 30000
#define KK 15
#define TT 400
#define NC 5
#define NB 12                 // basis rows actually used (1, t, 5x sin, 5x cos)
#define TWO_PI 6.283185307179586f
#define SMOOTH 0.2f

typedef __attribute__((ext_vector_type(2))) float v2f;
typedef __attribute__((ext_vector_type(4))) float v4f;
typedef __attribute__((ext_vector_type(8))) float v8f;

// ---------------------------------------------------------------------------
// Kernel 0: emd_signals[n,t] = sum_c emd_seasonal[n,c,t]
// emd_seasonal (192 MB) is read once -> non-temporal loads so it does not
// evict the emd_signals working set (48 MB, reused 16x) from the 192 MB L2.
// ---------------------------------------------------------------------------
__global__ __launch_bounds__(256) void emd_sum_kernel(const float* __restrict__ emd,
                                                      float* __restrict__ emd_sig) {
    int i = blockIdx.x * 256 + threadIdx.x;           // float4 index
    if (i >= NN * (TT / 4)) return;
    int n  = i / (TT / 4);
    int t4 = i - n * (TT / 4);
    const v4f* p = (const v4f*)emd + n * TT + t4;     // n*4*(TT/4) = n*TT
    v4f s = __builtin_nontemporal_load(p)
          + __builtin_nontemporal_load(p + (TT / 4))
          + __builtin_nontemporal_load(p + 2 * (TT / 4))
          + __builtin_nontemporal_load(p + 3 * (TT / 4));
    ((v4f*)emd_sig)[i] = s;                           // regular store: keep L2-resident
}

// ---------------------------------------------------------------------------
// Kernel 1: 16-wide coefficient rows per node:
//   col0=offset, col1=trend, col 2+c = amp_s*cos(ph_s), col 7+c = amp_s*sin(ph_s)
// where amp_s/ph_s are the SMOOTH-blended spatial averages (circular for phase).
// ---------------------------------------------------------------------------
__global__ __launch_bounds__(256) void coef_kernel(const float* __restrict__ coff,
                                                   const float* __restrict__ trend,
                                                   const int*   __restrict__ idx,
                                                   const float* __restrict__ nwt,
                                                   const float* __restrict__ amp,
                                                   const float* __restrict__ phs,
                                                   float* __restrict__ acoef) {
    int n = blockIdx.x * 256 + threadIdx.x;
    if (n >= NN) return;
    float sumA[NC], sS[NC], sC[NC];
#pragma unroll
    for (int c = 0; c < NC; ++c) { sumA[c] = 0.f; sS[c] = 0.f; sC[c] = 0.f; }
    for (int k = 0; k < KK; ++k) {
        int   j = idx[n * KK + k];
        float w = nwt[n * KK + k];
#pragma unroll
        for (int c = 0; c < NC; ++c) {
            float a  = amp[j * NC + c];
            float ph = phs[j * NC + c];
            sumA[c] += w * a;
            sS[c]   += w * sinf(ph);
            sC[c]   += w * cosf(ph);
        }
    }
    float* A = acoef + n * 16;
    A[0] = coff[n];
    A[1] = trend[n];
#pragma unroll
    for (int c = 0; c < NC; ++c) {
        float as = (1.f - SMOOTH) * amp[n * NC + c] + SMOOTH * sumA[c];
        float ps = (1.f - SMOOTH) * phs[n * NC + c] + SMOOTH * atan2f(sS[c], sC[c]);
        A[2 + c] = as * cosf(ps);   // coefficient of sin(w t)
        A[7 + c] = as * sinf(ps);   // coefficient of cos(w t)
    }
    A[12] = 0.f; A[13] = 0.f; A[14] = 0.f; A[15] = 0.f;
}

// ---------------------------------------------------------------------------
// Kernel 1b: basis matrix B[12][T] (19.2 KB, shared by all blocks; stays hot
// in cache). Row 0 = 1, row 1 = t, rows 2..6 = sin(w_c t), rows 7..11 = cos(w_c t).
// This removes ALL transcendentals from the hot fused kernel.
// ---------------------------------------------------------------------------
__global__ __launch_bounds__(256) void basis_kernel(const float* __restrict__ tvec,
                                                    const float* __restrict__ per,
                                                    float* __restrict__ basisB) {
    int i = blockIdx.x * 256 + threadIdx.x;           // i = r*T + t
    if (i >= NB * TT) return;
    int r = i / TT, t = i - r * TT;
    float tf = tvec[t];
    float v;
    if (r == 0) {
        v = 1.f;
    } else if (r == 1) {
        v = tf;
    } else if (r < 7) {
        float p = fminf(fmaxf(per[r - 2], 15.f), 350.f);
        v = sinf((TWO_PI / p) * tf);
    } else {
        float p = fminf(fmaxf(per[r - 7], 15.f), 350.f);
        v = cosf((TWO_PI / p) * tf);
    }
    basisB[i] = v;
}

// ---------------------------------------------------------------------------
// Kernel 2: block = 256 threads (8 waves) handles 16 nodes x all T.
// Per 16x16 (node x time) tile per wave:
//   1) 6 coalesced loads of the precomputed basis tile (cache-resident)
//   2) 3x V_WMMA_F32_16X16X4_F32 for linear + harmonic part
//   3) spill accumulator to per-wave LDS tile (stride 17)
//   4) gather phase: rolled r-loop, 15 batched 32-bit-index L2 gathers per
//      node row, blend, add C from LDS, coalesced non-temporal store.
// ---------------------------------------------------------------------------
__global__ __launch_bounds__(256) void fused_signal_kernel(
        const float* __restrict__ emd_sig,
        const float* __restrict__ acoef,
        const float* __restrict__ basisB,
        const int*   __restrict__ idx,
        const float* __restrict__ nwt,
        const float* __restrict__ lwt,
        const float* __restrict__ esw,
        const float* __restrict__ lsw,
        float* __restrict__ out) {
    __shared__ int   s_jw[16 * 16];    // neighbor row word-offset: idx*TT (stride 16)
    __shared__ float s_nw[16 * 16];
    __shared__ float s_lw[16 * 16];
    __shared__ float s_ac[16 * 16];
    __shared__ float s_ew[16];
    __shared__ float s_lws[16];
    __shared__ float s_c[8 * 16 * 17]; // per-wave C tile, row stride 17

    const int tid = threadIdx.x;
    const int nb  = blockIdx.x * 16;              // node base (N % 16 == 0)

    if (tid < 16 * KK) {
        int m = tid / KK, k = tid - m * KK;
        int g = (nb + m) * KK + k;
        s_jw[m * 16 + k] = idx[g] * TT;           // 32-bit word offset (N*T < 2^31)
        s_nw[m * 16 + k] = nwt[g];
        s_lw[m * 16 + k] = lwt[g];
    }
    s_ac[tid] = acoef[nb * 16 + tid];
    if (tid < 16) {
        s_ew[tid]  = 1.f / (1.f + expf(-esw[nb + tid]));
        s_lws[tid] = 1.f / (1.f + expf(-lsw[nb + tid]));
    }
    __syncthreads();

    const int lane = tid & 31;
    const int wave = tid >> 5;
    const int hi   = lane >> 4;                   // 0 / 1 half-wave
    const int lm   = lane & 15;
    float* cbase = s_c + wave * (16 * 17);

    // A matrix (16x4 f32 per WMMA): lanes 0-15 K={4j,4j+1}, lanes 16-31 K={4j+2,4j+3}
    v2f A0, A1, A2;
    A0.x = s_ac[lm * 16 + 0 + 2 * hi];  A0.y = s_ac[lm * 16 + 1 + 2 * hi];
    A1.x = s_ac[lm * 16 + 4 + 2 * hi];  A1.y = s_ac[lm * 16 + 5 + 2 * hi];
    A2.x = s_ac[lm * 16 + 8 + 2 * hi];  A2.y = s_ac[lm * 16 + 9 + 2 * hi];

    // 25 time tiles of 16, strided across the 8 waves
    for (int tt = wave; tt < TT / 16; tt += 8) {
        const int t = tt * 16 + lm;

        // B matrix (4x16 f32): lanes 0-15 K={4j,4j+1}, lanes 16-31 K={4j+2,4j+3};
        // N = lane&15. Values come from the precomputed cache-resident basis table.
        const float* Bp = basisB + t + 2 * hi * TT;
        v2f B0, B1, B2;
        B0.x = Bp[0 * TT];  B0.y = Bp[1 * TT];
        B1.x = Bp[4 * TT];  B1.y = Bp[5 * TT];
        B2.x = Bp[8 * TT];  B2.y = Bp[9 * TT];

        v8f c = {};
        c = __builtin_amdgcn_wmma_f32_16x16x4_f32(false, A0, false, B0, (short)0, c, false, false);
        c = __builtin_amdgcn_wmma_f32_16x16x4_f32(false, A1, false, B1, (short)0, c, false, false);
        c = __builtin_amdgcn_wmma_f32_16x16x4_f32(false, A2, false, B2, (short)0, c, false, false);

        // Spill C to LDS: VGPR r, lanes 0-15 -> M=r, lanes 16-31 -> M=r+8; N = lane&15
#pragma unroll
        for (int r = 0; r < 8; ++r)
            cbase[(r + 8 * hi) * 17 + lm] = c[r];

        // Gather phase: rolled over the 8 node rows this lane contributes to.
#pragma unroll 2
        for (int r = 0; r < 8; ++r) {
            const int m = r + 8 * hi;
            float sg[KK];
#pragma unroll
            for (int k = 0; k < KK; ++k)
                sg[k] = emd_sig[s_jw[m * 16 + k] + t];   // batched, L2-resident
            float loc = 0.f, reg = 0.f;
#pragma unroll
            for (int k = 0; k < KK; ++k) {
                loc += s_lw[m * 16 + k] * sg[k];
                reg += s_nw[m * 16 + k] * sg[k];
            }
            const float emdv = emd_sig[(nb + m) * TT + t];
            const float ew = s_ew[m], lv = s_lws[m];
            const float adj = (1.f - ew) * emdv + ew * lv * loc + ew * (1.f - lv) * reg;
            const float val = cbase[m * 17 + lm] + adj;
            __builtin_nontemporal_store(val, &out[(nb + m) * TT + t]);  // coalesced 64B/half-wave
        }
    }
}

// ---------------------------------------------------------------------------
extern "C" void kernel_launch(void* const* d_in, const int* in_sizes, int n_in,
                              void* d_out, int out_size, void* d_ws, size_t ws_size,
                              hipStream_t stream) {
    const float* tvec  = (const float*)d_in[0];
    const float* coff  = (const float*)d_in[1];
    const float* trend = (const float*)d_in[2];
    const float* emd   = (const float*)d_in[3];
    const int*   idx   = (const int*)  d_in[4];
    const float* nwt   = (const float*)d_in[5];
    const float* lwt   = (const float*)d_in[6];
    const float* amp   = (const float*)d_in[7];
    const float* phs   = (const float*)d_in[8];
    const float* per   = (const float*)d_in[9];
    const float* esw   = (const float*)d_in[10];
    const float* lsw   = (const float*)d_in[11];
    float* out = (float*)d_out;

    // Workspace: emd_signals [N*T] | acoef [N*16] | basisB [12*T]   (~50 MB)
    float* emd_sig = (float*)d_ws;
    float* acoef   = emd_sig + (size_t)NN * TT;
    float* basisB  = acoef + (size_t)NN * 16;

    emd_sum_kernel<<<(NN * (TT / 4) + 255) / 256, 256, 0, stream>>>(emd, emd_sig);
    coef_kernel<<<(NN + 255) / 256, 256, 0, stream>>>(coff, trend, idx, nwt, amp, phs, acoef);
    basis_kernel<<<(NB * TT + 255) / 256, 256, 0, stream>>>(tvec, per, basisB);
    fused_signal_kernel<<<NN / 16, 256, 0, stream>>>(emd_sig, acoef, basisB, idx, nwt, lwt,
                                                     esw, lsw, out);
}